// quantile_activation_2d_65240553226665
// MI455X (gfx1250) — compile-verified
//
#include <hip/hip_runtime.h>
#include <hip/hip_bf16.h>

#define CC       256      // channels
#define NSAMP    1000     // context samples per channel
#define NTOT     1002     // + two sentinels
#define NPAD     1024     // power-of-two padded length for sort/search
#define NQ       100      // quantiles
#define QSTRIDE  128      // padded table stride (floats)
#define HW       3136     // 56*56
#define HW4      784      // HW/4
#define BN_EPS   1e-5f

// ======================= Kernel A =========================================
// Per channel: sort context row (+sentinels, +INF pad) with a bitonic sort in
// LDS, build closed-form cumulative weights (two linear ramps), then compute
// the 100 weighted quantiles (thresholds, y-domain) and the bn2-fused output
// LUT. Tables are padded to 128 entries (+INF thresholds) so the map kernel
// can use a fixed 7-step branchless binary search.
__global__ __launch_bounds__(512)
void quantile_prep_kernel(const float* __restrict__ ctx,
                          const float* __restrict__ g2,
                          const float* __restrict__ b2,
                          const float* __restrict__ m2,
                          const float* __restrict__ v2,
                          float* __restrict__ qtab,
                          float* __restrict__ ttab)
{
    __shared__ float sv[NPAD];
    __shared__ float cw[NPAD];
    __shared__ int   s_sn;

    const int c   = blockIdx.x;
    const int tid = threadIdx.x;

    for (int i = tid; i < NPAD; i += 512) {
        float val;
        if (i < NSAMP)           val = ctx[c * NSAMP + i];
        else if (i == NSAMP)     val = -100.0f;      // negative sentinel
        else if (i == NSAMP + 1) val =  100.0f;      // positive sentinel
        else                     val = __builtin_inff();  // sort-to-end pad
        sv[i] = val;
    }
    __syncthreads();

    // Bitonic sort, ascending, 1024 elements, 512 threads (2 pairs/thread).
    for (int k = 2; k <= NPAD; k <<= 1) {
        for (int j = k >> 1; j > 0; j >>= 1) {
            for (int t0 = tid; t0 < NPAD; t0 += 512) {
                int ixj = t0 ^ j;
                if (ixj > t0) {
                    float a = sv[t0];
                    float b = sv[ixj];
                    bool up = ((t0 & k) == 0);
                    if ((a > b) == up) { sv[t0] = b; sv[ixj] = a; }
                }
            }
            __syncthreads();
        }
    }

    // sn = count of strictly-negative values (they sort first).
    if (tid == 0) {
        int pos = 0;
        #pragma unroll
        for (int step = 512; step >= 1; step >>= 1)
            if (sv[pos + step - 1] < 0.0f) pos += step;
        s_sn = pos;
    }
    __syncthreads();

    const int   sn = s_sn;
    const int   sp = NTOT - sn;
    // reference: w = (1/sp) then *1000 (cond is always true thanks to sentinels)
    const float wn = __fmul_rn(__fdiv_rn(1.0f, (float)sn), 1000.0f);
    const float wp = __fmul_rn(__fdiv_rn(1.0f, (float)sp), 1000.0f);
    const float negtot = __fmul_rn(wn, (float)sn);

    for (int i = tid; i < NPAD; i += 512) {
        float cv;
        if (i < sn)        cv = __fmul_rn(wn, (float)(i + 1));
        else if (i < NTOT) cv = __fadd_rn(negtot, __fmul_rn(wp, (float)(i + 1 - sn)));
        else               cv = __builtin_inff();
        cw[i] = cv;
    }
    __syncthreads();

    if (tid < QSTRIDE) {
        if (tid < NQ) {
            const float total = cw[NTOT - 1];
            const float qk = (float)(tid + 1) * (1.0f / 101.0f);   // quant_list[k]
            const float tw = qk * total;
            // searchsorted(cw, tw, 'right') == count of cw <= tw
            int pos = 0;
            #pragma unroll
            for (int step = 512; step >= 1; step >>= 1)
                if (cw[pos + step - 1] <= tw) pos += step;
            const int above = min(pos, NSAMP - 1);                 // clip to 999
            const int below = min(max(pos - 1, 0), NSAMP - 1);
            const float wbv = cw[below], wav = cw[above];
            const float vb  = sv[below], va  = sv[above];
            const float frac = (tw - wbv) / (wav - wbv + 1e-6f);
            qtab[c * QSTRIDE + tid] = vb + (va - vb) * frac;

            // bn2 folded into the output LUT: T[k] = (q_list[k]-mean2)*scale2+beta2
            const float sc2 = __fdiv_rn(g2[c], __fsqrt_rn(__fadd_rn(v2[c], BN_EPS)));
            ttab[c * QSTRIDE + tid] =
                __fadd_rn(__fmul_rn(__fsub_rn(qk, m2[c]), sc2), b2[c]);
        } else {
            qtab[c * QSTRIDE + tid] = __builtin_inff();  // pad: never counted
            ttab[c * QSTRIDE + tid] = 0.0f;              // pad: never read (idx clamped)
        }
    }
}

// ======================= Kernel B =========================================
// One block per contiguous (b,c) slab of HW floats. Stage the channel's
// threshold table + output LUT into LDS via CDNA5 async global->LDS loads
// (ASYNCcnt path), then a float4-vectorized bn1 + 7-step branchless
// binary search + LUT gather. Purely HBM-bandwidth-bound.
__global__ __launch_bounds__(256)
void qmap_kernel(const float* __restrict__ x,
                 const float* __restrict__ qtab,
                 const float* __restrict__ ttab,
                 const float* __restrict__ g1,
                 const float* __restrict__ b1,
                 const float* __restrict__ m1,
                 const float* __restrict__ v1,
                 float* __restrict__ out)
{
    __shared__ float q_lds[QSTRIDE];
    __shared__ float t_lds[QSTRIDE];

    const int slab = blockIdx.x;
    const int c    = slab & (CC - 1);   // layout (B,C,H,W): slab = b*C + c
    const int tid  = threadIdx.x;

    // Wave 0: 32 lanes x b128 = 512B per table, async into LDS.
    if (tid < 32) {
        const float* gq = qtab + (size_t)c * QSTRIDE + tid * 4;
        const float* gt = ttab + (size_t)c * QSTRIDE + tid * 4;
        // low 32 bits of a flat LDS pointer == LDS byte offset (ISA 10.2)
        unsigned lq = (unsigned)(size_t)(&q_lds[tid * 4]);
        unsigned lt = (unsigned)(size_t)(&t_lds[tid * 4]);
        asm volatile("global_load_async_to_lds_b128 %0, %1, off"
                     :: "v"(lq), "v"(gq) : "memory");
        asm volatile("global_load_async_to_lds_b128 %0, %1, off"
                     :: "v"(lt), "v"(gt) : "memory");
        asm volatile("s_wait_asynccnt 0x0" ::: "memory");
    }
    __syncthreads();

    const float sc1 = __fdiv_rn(g1[c], __fsqrt_rn(__fadd_rn(v1[c], BN_EPS)));
    const float mm1 = m1[c];
    const float bb1 = b1[c];

    const float4* __restrict__ xin = (const float4*)(x   + (size_t)slab * HW);
    float4*       __restrict__ oo  = (float4*)      (out + (size_t)slab * HW);

    for (int i = tid; i < HW4; i += 256) {
        float4 v = xin[i];
        float ys[4] = { v.x, v.y, v.z, v.w };
        float rs[4];
        #pragma unroll
        for (int l = 0; l < 4; ++l) {
            // y = (x - mean1)*scale1 + beta1  (same op order as reference)
            float y = __fadd_rn(__fmul_rn(__fsub_rn(ys[l], mm1), sc1), bb1);
            // count of thresholds <= y over the 128-padded (+INF) table
            int pos = 0;
            #pragma unroll
            for (int step = 64; step >= 1; step >>= 1)
                if (q_lds[pos + step - 1] <= y) pos += step;
            rs[l] = t_lds[min(pos, NQ - 1)];
        }
        float4 r; r.x = rs[0]; r.y = rs[1]; r.z = rs[2]; r.w = rs[3];
        oo[i] = r;
    }
}

extern "C" void kernel_launch(void* const* d_in, const int* in_sizes, int n_in,
                              void* d_out, int out_size, void* d_ws, size_t ws_size,
                              hipStream_t stream) {
    (void)in_sizes; (void)n_in; (void)ws_size;
    const float* x   = (const float*)d_in[0];
    const float* ctx = (const float*)d_in[1];
    const float* g1  = (const float*)d_in[2];
    const float* b1  = (const float*)d_in[3];
    const float* m1  = (const float*)d_in[4];
    const float* v1  = (const float*)d_in[5];
    const float* g2  = (const float*)d_in[6];
    const float* b2  = (const float*)d_in[7];
    const float* m2  = (const float*)d_in[8];
    const float* v2  = (const float*)d_in[9];
    float* out = (float*)d_out;

    // workspace: thresholds + fused-bn2 LUT, 256*128 floats each (262144 B)
    float* qtab = (float*)d_ws;
    float* ttab = qtab + CC * QSTRIDE;

    quantile_prep_kernel<<<CC, 512, 0, stream>>>(ctx, g2, b2, m2, v2, qtab, ttab);

    const int nslab = out_size / HW;   // B*C = 4096
    qmap_kernel<<<nslab, 256, 0, stream>>>(x, qtab, ttab, g1, b1, m1, v1, out);
}